// MultiLIF_17059610100016
// MI455X (gfx1250) — compile-verified
//
#include <hip/hip_runtime.h>
#include <cstdint>
#include <cstddef>

// LIF spiking-neuron scan, forward pass.
//   I : [B=32, L=2048, K=1024] f32
//   out (concat): spikes[B,L,K] | series[B,L,K] | v_seq[B,L,K]
//
// 32768 independent length-2048 recurrences (one per (b,k) column).
// CDNA5 path: 7-deep ring of global_load_async_to_lds_b32 chunk prefetches
// (56 outstanding async ops, under the 6-bit ASYNCcnt limit of 63),
// s_wait_asynccnt pipelining, non-temporal load/store cache policy.
//
// Roofline: 268 MB read + 805 MB written -> ~46 us at 23.3 TB/s; compute is
// trivial, so the design goal is pure latency hiding on the input stream.

namespace {
constexpr int Bb    = 32;
constexpr int Ll    = 2048;
constexpr int Kk    = 1024;
constexpr int TPB   = 256;             // 8 wave32 per block
constexpr int CHUNK = 8;               // timesteps per async batch
constexpr int NBUF  = 7;               // ring depth: 7*8 = 56 outstanding max (<= 63)
constexpr int NCHUNK = Ll / CHUNK;     // 256
constexpr int BLK_PER_ROW = Kk / TPB;  // 4 blocks per batch row -> 128 blocks
static_assert((NBUF - 1) * CHUNK == 48, "steady-state wait literal below is 0x30");
static_assert(NBUF * CHUNK <= 63, "ASYNCcnt is a 6-bit counter");
}

__global__ __launch_bounds__(TPB)
void lif_scan_kernel(const float* __restrict__ gI,
                     float* __restrict__ o_spk,
                     float* __restrict__ o_ser,
                     float* __restrict__ o_v)
{
    // 7-buffer ring: 7 * 8 * 256 * 4B = 56 KB of the 320 KB WGP LDS.
    // tid-major rows: async writes and ds_load_b32 reads are bank-conflict-free.
    __shared__ float smem[NBUF][CHUNK][TPB];

    const int tid = (int)threadIdx.x;
    const int b   = (int)blockIdx.x / BLK_PER_ROW;
    const int k0  = ((int)blockIdx.x % BLK_PER_ROW) * TPB;

    // uniform (SGPR) global base for this block's (b, k-range) column panel
    const unsigned long long gbase =
        (unsigned long long)(uintptr_t)(gI + (size_t)b * Ll * Kk + k0);

    // raw 32-bit LDS offset of this thread's slot (low 32 bits of flat LDS ptr)
    const unsigned lbase = (unsigned)(uintptr_t)(void*)&smem[0][0][tid];

    // Issue CHUNK async global->LDS copies of timestep-chunk c into ring slot buf.
    // Each lane copies exactly the elements it will itself consume, so the
    // per-wave ASYNCcnt is the only synchronization needed (no barriers).
    auto issue = [&](int c, int buf) {
        unsigned laddr = lbase + (unsigned)(buf * CHUNK * TPB * 4);
        int voff = (c * CHUNK * Kk + tid) * 4;
#pragma unroll
        for (int tt = 0; tt < CHUNK; ++tt) {
            asm volatile("global_load_async_to_lds_b32 %0, %1, %2 th:TH_LOAD_NT"
                         :: "v"(laddr), "v"(voff), "s"(gbase)
                         : "memory");
            laddr += TPB * 4;
            voff  += Kk * 4;
        }
    };

    float v = 0.0f, a = 0.0f, snum = 0.0f;
    size_t oidx = (size_t)b * Ll * Kk + k0 + tid;

    // One chunk of the recurrence, reading staged inputs from ring slot buf.
    auto compute = [&](int buf) {
#pragma unroll
        for (int tt = 0; tt < CHUNK; ++tt) {
            const float It = smem[buf][tt][tid];

            // membrane update (pre-reset):  v += -v/TAU + I_t
            v = v + (It - v * (1.0f / 20.0f));

            // th_eff = TH + BETA_ADAPT * a
            const float th_eff = __builtin_fmaf(1.5f, a, 1.5f);

            // Straight-through forward value == hard spike, exactly in fp32:
            //   s_hard=0: c + (0-c) == 0;  s_hard=1: v>=th_eff => s_soft in [0.5,1]
            //   => (1-c) exact (Sterbenz) => c + (1-c) == 1 exactly.
            const float s = (v >= th_eff) ? 1.0f : 0.0f;
            snum += s;

            // streaming outputs (write set = 805 MB >> 192 MB L2 -> non-temporal)
            __builtin_nontemporal_store(s,    o_spk + oidx);
            __builtin_nontemporal_store(snum, o_ser + oidx);
            __builtin_nontemporal_store(v,    o_v   + oidx);

            // reset (exact select since s in {0,1}) and adaptation update
            v = (s != 0.0f) ? -0.5f : v;
            a = a + (s - a * (1.0f / 100.0f));

            oidx += Kk;
        }
    };

    // Prologue: fill the ring (56 async loads in flight).
#pragma unroll
    for (int c = 0; c < NBUF; ++c) issue(c, c);

    // Steady state: wait for oldest chunk (<= (NBUF-1)*CHUNK = 48 outstanding),
    // compute it, then recycle its slot for chunk c+NBUF.
    int rd = 0;
    for (int c = 0; c < NCHUNK - NBUF; ++c) {
        asm volatile("s_wait_asynccnt 0x30" ::: "memory");
        compute(rd);
        issue(c + NBUF, rd);
        rd = (rd + 1 == NBUF) ? 0 : rd + 1;
    }

    // Drain: everything resident, compute the last NBUF chunks back-to-back.
    asm volatile("s_wait_asynccnt 0x0" ::: "memory");
    for (int c = 0; c < NBUF; ++c) {
        compute(rd);
        rd = (rd + 1 == NBUF) ? 0 : rd + 1;
    }
}

extern "C" void kernel_launch(void* const* d_in, const int* in_sizes, int n_in,
                              void* d_out, int out_size, void* d_ws, size_t ws_size,
                              hipStream_t stream) {
    (void)in_sizes; (void)n_in; (void)out_size; (void)d_ws; (void)ws_size;
    const float* I  = (const float*)d_in[0];
    float* out      = (float*)d_out;
    const size_t N  = (size_t)Bb * Ll * Kk;          // 67,108,864 per output tensor
    dim3 grid(Bb * BLK_PER_ROW);                     // 128 blocks
    dim3 block(TPB);                                 // 256 threads = 8 wave32
    lif_scan_kernel<<<grid, block, 0, stream>>>(I, out, out + N, out + 2 * N);
}